// CombinedSPALoss_62242666053902
// MI455X (gfx1250) — compile-verified
//
#include <hip/hip_runtime.h>
#include <hip/hip_bf16.h>

// CDNA5 / gfx1250: wave32, WMMA 16x16x4 f32.
typedef __attribute__((ext_vector_type(2))) float v2f;
typedef __attribute__((ext_vector_type(8))) float v8f;

#define C_DIM 256
#define ROWS_PER_BLOCK 16
#define EPS_F 1e-8f

// ---------------------------------------------------------------------------
// zero the global accumulators (harness does not re-poison between replays)
// ws layout: [0] double spa_sum, [1] double pair_sum, [2] u32 pair_count
// ---------------------------------------------------------------------------
__global__ void spa_zero_kernel(double* spa, double* pair, unsigned int* cnt) {
    *spa = 0.0;
    *pair = 0.0;
    *cnt = 0u;
}

// ---------------------------------------------------------------------------
// Main streaming kernel: 16 rows per block, 256 threads (8 wave32s).
// thread t handles row r = t/16, columns (t%16) + 16k, k=0..15 (coalesced).
// Per-row: 16 pos-moments v_j = sum t*q^j, 16 neg-moments u_j = sum (1-t)*q^j.
// pair_sum_row = u^T W v  with W[j][k] = a_{j+k} * C(j+k,j) * (-1)^k,
// a_d = Taylor coefficients of softplus at 0 (exact rationals, degree 14,
// truncation error <= 1.4e-9 on |diff|<1, i.e. below f32 eps).
// The 16-row batched bilinear runs on V_WMMA_F32_16X16X4_F32 (exact f32).
// ---------------------------------------------------------------------------
__global__ __launch_bounds__(256) void spa_lpr_kernel(
    const float* __restrict__ logits, const float* __restrict__ targets,
    double* __restrict__ spa_sum, double* __restrict__ pair_sum,
    unsigned int* __restrict__ pair_count) {
    __shared__ float Wm[16][16];   // polynomial bilinear matrix
    __shared__ float Ul[16][16];   // neg moments, row-major [row][moment]
    __shared__ float Vl[16][16];   // pos moments
    __shared__ double spa_block[8];

    const int tid = threadIdx.x;

    // ---- build W in LDS (one entry per thread; exact rational coefficients) ----
    {
        const double acoef[15] = {
            0.6931471805599453,          // a0  = ln 2
            0.5,                         // a1
            1.0 / 8.0,                   // a2
            0.0,
            -1.0 / 192.0,                // a4
            0.0,
            1.0 / 2880.0,                // a6
            0.0,
            -17.0 / 645120.0,            // a8
            0.0,
            31.0 / 14515200.0,           // a10
            0.0,
            -691.0 / 3832012800.0,       // a12
            0.0,
            10922.0 / 697426329600.0     // a14
        };
        const int j = tid >> 4, k = tid & 15;
        const int d = j + k;
        float w = 0.0f;
        if (d <= 14) {
            const double ad = acoef[d];
            if (ad != 0.0) {
                double b = 1.0;  // C(d, j)
                for (int i = 1; i <= j; ++i) b = b * (double)(d - j + i) / (double)i;
                w = (float)(ad * b * ((k & 1) ? -1.0 : 1.0));
            }
        }
        Wm[j][k] = w;
    }

    // ---- streaming phase: sigmoid + BCE + moment accumulation ----
    const int r = tid >> 4;   // row within block (0..15)
    const int ci = tid & 15;  // column phase
    const long long row = (long long)blockIdx.x * ROWS_PER_BLOCK + r;
    const float* __restrict__ lrow = logits + row * C_DIM;
    const float* __restrict__ trow = targets + row * C_DIM;

    float u[16], v[16];
#pragma unroll
    for (int j = 0; j < 16; ++j) { u[j] = 0.0f; v[j] = 0.0f; }
    float spa = 0.0f;

#pragma unroll
    for (int kk = 0; kk < 16; ++kk) {
        const int c = ci + (kk << 4);
        const float x = lrow[c];
        const float t = trow[c];            // exactly 0.0 or 1.0
        const float q = 1.0f / (1.0f + expf(-x));
        spa += -t * logf(q + EPS_F) - (1.0f - t) * logf(1.0f - q + EPS_F);
        const float posw = t, negw = 1.0f - t;
        float pw = 1.0f;
#pragma unroll
        for (int j = 0; j < 16; ++j) {
            u[j] += negw * pw;
            v[j] += posw * pw;
            pw *= q;
        }
    }

    // ---- per-row reduction across the 16 lanes owning this row ----
#pragma unroll
    for (int m = 8; m >= 1; m >>= 1) {
#pragma unroll
        for (int j = 0; j < 16; ++j) {
            u[j] += __shfl_xor(u[j], m, 16);
            v[j] += __shfl_xor(v[j], m, 16);
        }
    }
    if (ci == 0) {
#pragma unroll
        for (int j = 0; j < 16; ++j) { Ul[r][j] = u[j]; Vl[r][j] = v[j]; }
        // exact pair count: P_i * N_i (small integers, exact in f32)
        atomicAdd(pair_count, (unsigned int)(u[0] * v[0] + 0.5f));
    }

    // ---- SPA (BCE) block reduction ----
#pragma unroll
    for (int m = 16; m >= 1; m >>= 1) spa += __shfl_xor(spa, m, 32);
    if ((tid & 31) == 0) spa_block[tid >> 5] = (double)spa;
    __syncthreads();
    if (tid == 0) {
        double s = 0.0;
#pragma unroll
        for (int w = 0; w < 8; ++w) s += spa_block[w];
        atomicAdd(spa_sum, s);
    }

    // ---- bilinear u^T W v for all 16 rows via WMMA f32 16x16x4 (wave 0) ----
    if (tid < 32) {  // wave-uniform branch: EXEC all-ones inside (WMMA reqmt)
        const int lane = tid;
        const int hi = lane >> 4;  // 0: lanes 0-15, 1: lanes 16-31
        const int lm = lane & 15;
        v8f acc = {};  // C = 0
#pragma unroll
        for (int c4 = 0; c4 < 4; ++c4) {
            // A (16x4 f32) layout: lanes 0-15 hold K=0,1; lanes 16-31 hold K=2,3
            const int ka = 4 * c4 + 2 * hi;
            v2f a, b;
            a.x = Ul[lm][ka];
            a.y = Ul[lm][ka + 1];
            // B (4x16 f32): mirrored striping, rows ka/ka+1 across lanes
            b.x = Wm[ka][lm];
            b.y = Wm[ka + 1][lm];
            acc = __builtin_amdgcn_wmma_f32_16x16x4_f32(
                false, a, false, b, (short)0, acc, false, false);
        }
        // D layout: VGPR rr of lane L = M1[rr + 8*hi][lm]; dot with V per row.
        float part = 0.0f;
#pragma unroll
        for (int rr = 0; rr < 8; ++rr) part += acc[rr] * Vl[rr + 8 * hi][lm];
        // reduce over the 16 columns within each half, then combine halves
#pragma unroll
        for (int m = 8; m >= 1; m >>= 1) part += __shfl_xor(part, m, 16);
        part += __shfl_xor(part, 16, 32);
        if (lane == 0) atomicAdd(pair_sum, (double)part);
    }
}

// ---------------------------------------------------------------------------
// finalize: out = spa_mean + 0.1 * pair_sum / (count + eps)
// ---------------------------------------------------------------------------
__global__ void spa_finalize_kernel(const double* spa, const double* pair,
                                    const unsigned int* cnt, float* out,
                                    double inv_elems) {
    const double lpr = *pair / ((double)(*cnt) + 1e-8);
    out[0] = (float)(*spa * inv_elems + 0.1 * lpr);
}

extern "C" void kernel_launch(void* const* d_in, const int* in_sizes, int n_in,
                              void* d_out, int out_size, void* d_ws, size_t ws_size,
                              hipStream_t stream) {
    (void)n_in; (void)out_size; (void)ws_size;
    const float* logits = (const float*)d_in[0];
    const float* targets = (const float*)d_in[1];
    const int total = in_sizes[0];      // B * 256
    const int B = total / C_DIM;        // 1024

    double* spa = (double*)d_ws;
    double* pair = spa + 1;
    unsigned int* cnt = (unsigned int*)(pair + 1);

    spa_zero_kernel<<<1, 1, 0, stream>>>(spa, pair, cnt);
    spa_lpr_kernel<<<B / ROWS_PER_BLOCK, 256, 0, stream>>>(logits, targets, spa,
                                                           pair, cnt);
    spa_finalize_kernel<<<1, 1, 0, stream>>>(spa, pair, cnt, (float*)d_out,
                                             1.0 / (double)total);
}